// KmeansNN_11665131176018
// MI455X (gfx1250) — compile-verified
//
#include <hip/hip_runtime.h>

typedef __attribute__((ext_vector_type(2))) float v2f;
typedef __attribute__((ext_vector_type(8))) float v8f;
typedef int v4i __attribute__((vector_size(16)));   // matches builtin param pointee

#define B_   16384
#define K_   2048
#define D_   128
#define KCH  64      // centers staged per LDS chunk (= 4 N-tiles of 16)
#define PITCH 132    // LDS row pitch in floats (conflict-free for lane-strided reads)

#if defined(__AMDGCN__) && __has_builtin(__builtin_amdgcn_global_load_async_to_lds_b128) && \
    __has_builtin(__builtin_amdgcn_s_wait_asynccnt)
#define USE_ASYNC_LDS 1
#else
#define USE_ASYNC_LDS 0
#endif

// ---------------- Kernel 1: c2[k] = ||center_k||^2, and copy center into d_out ----
__global__ __launch_bounds__(256) void c2_kernel(const float* __restrict__ center,
                                                 float* __restrict__ c2,
                                                 float* __restrict__ center_copy) {
  int tid  = threadIdx.x;
  int wave = tid >> 5, lane = tid & 31;
  int row  = blockIdx.x * 8 + wave;                 // one wave32 per center row
  const float4* src = reinterpret_cast<const float4*>(center + (size_t)row * D_);
  float4 v = src[lane];                             // 32 lanes x float4 = 128 floats
  reinterpret_cast<float4*>(center_copy + (size_t)row * D_)[lane] = v;
  float s = v.x*v.x + v.y*v.y + v.z*v.z + v.w*v.w;
  #pragma unroll
  for (int off = 16; off; off >>= 1) s += __shfl_xor(s, off, 32);
  if (lane == 0) c2[row] = s;
}

// ---------------- Kernel 2: fused  x@C^T -> dist -> online softmax/argmax -> gather
__global__ __launch_bounds__(256) void knn_kernel(const float* __restrict__ x,
                                                  const float* __restrict__ center,
                                                  const float* __restrict__ c2buf,
                                                  float* __restrict__ out,
                                                  int* __restrict__ labels) {
#if USE_ASYNC_LDS
  __shared__ float ct[2][KCH * PITCH];   // ping-pong center chunks
#else
  __shared__ float ct[1][KCH * PITCH];
#endif
  __shared__ int   s_label[128];
  __shared__ float s_scale[128];

  const int tid  = threadIdx.x;
  const int wave = tid >> 5;
  const int lane = tid & 31;
  const int hi   = lane >> 4;         // half-wave selects K-pair (+2) per ISA layout
  const int lq   = lane & 15;
  const int m0   = blockIdx.x * 128;  // 128 B-rows per workgroup
  const int rowA = m0 + wave * 16 + lq;

  // ---- A operand: 16 rows x 128 K, f32 WMMA A-layout.
  // VGPR pair j holds K = 4j..4j+3:  lanes 0-15 -> (4j,4j+1), lanes 16-31 -> (4j+2,4j+3)
  v2f a[32];
  const v2f* px = reinterpret_cast<const v2f*>(x + (size_t)rowA * D_);
  #pragma unroll
  for (int j = 0; j < 32; ++j) a[j] = px[2 * j + hi];

  // ---- x2 per row (each half-lane holds half the K range; combine across halves)
  float psum = 0.f;
  #pragma unroll
  for (int j = 0; j < 32; ++j) psum += a[j].x * a[j].x + a[j].y * a[j].y;
  float x2full = psum + __shfl_xor(psum, 16, 32);   // full ||x_row||^2, row = lq
  float x2v[8];
  #pragma unroll
  for (int v = 0; v < 8; ++v) x2v[v] = __shfl(x2full, hi * 8 + v, 32);

  // ---- per-lane online state: 8 rows x (best logit, best k, sum exp)
  float bestl[8], esum[8]; int bestk[8];
  #pragma unroll
  for (int v = 0; v < 8; ++v) { bestl[v] = -1e30f; esum[v] = 0.f; bestk[v] = 0; }

#if USE_ASYNC_LDS
  // issue a whole chunk (64 rows x 128 f32) as 8 async b128 loads per thread
  auto issue_chunk = [&](int kb, int buf) {
    #pragma unroll
    for (int i = 0; i < 8; ++i) {
      int f = tid + 256 * i;
      int r = f >> 5, d4 = f & 31;
      const float* g = center + (size_t)(kb + r) * D_ + d4 * 4;
      float*       l = &ct[buf][r * PITCH + d4 * 4];
      __builtin_amdgcn_global_load_async_to_lds_b128(
          (__attribute__((address_space(1))) v4i*)g,
          (__attribute__((address_space(3))) v4i*)l, 0, 0);
    }
  };
  issue_chunk(0, 0);
#endif

  for (int kb = 0; kb < K_; kb += KCH) {
    const int  bufi     = (kb / KCH) & 1;
    const bool has_next = (kb + KCH) < K_;

#if USE_ASYNC_LDS
    // next chunk DMA into the idle buffer; overlaps this chunk's 128 WMMAs.
    if (has_next) issue_chunk(kb + KCH, bufi ^ 1);
    if (has_next) __builtin_amdgcn_s_wait_asynccnt(8);   // current chunk landed
    else          __builtin_amdgcn_s_wait_asynccnt(0);
    __syncthreads();                                     // visible to all waves
#else
    // synchronous staging (transient regs, no persistent live range)
    {
      #pragma unroll
      for (int i = 0; i < 8; ++i) {
        int f = tid + 256 * i;
        float4 vv = reinterpret_cast<const float4*>(center)[(size_t)(kb + (f >> 5)) * 32 + (f & 31)];
        *reinterpret_cast<float4*>(&ct[0][(f >> 5) * PITCH + (f & 31) * 4]) = vv;
      }
      if (has_next)
        __builtin_prefetch(center + (size_t)(kb + KCH) * D_ + tid * 32, 0, 1);
      __syncthreads();
    }
#endif

    // ---- 4 N-tiles concurrently: 4 independent accumulator chains
    float c2n[4];
    #pragma unroll
    for (int t = 0; t < 4; ++t) c2n[t] = c2buf[kb + t * 16 + lq];

    v8f acc[4];
    #pragma unroll
    for (int t = 0; t < 4; ++t) acc[t] = (v8f){};

    const float* cbase = &ct[0][0] + bufi * (KCH * PITCH) * (USE_ASYNC_LDS ? 1 : 0);
    #pragma unroll
    for (int j = 0; j < 32; ++j) {
      v2f b[4];
      #pragma unroll
      for (int t = 0; t < 4; ++t)
        b[t] = *reinterpret_cast<const v2f*>(
            &cbase[(t * 16 + lq) * PITCH + 4 * j + 2 * hi]);
      #pragma unroll
      for (int t = 0; t < 4; ++t)
        acc[t] = __builtin_amdgcn_wmma_f32_16x16x4_f32(false, a[j], false, b[t],
                                                       (short)0, acc[t], false, false);
    }

    // ---- epilogue: dist -> logit -> online argmax + exp-sum
    #pragma unroll
    for (int t = 0; t < 4; ++t) {
      const int k = kb + t * 16 + lq;     // this lane's column (center index)
      #pragma unroll
      for (int v = 0; v < 8; ++v) {
        float d2 = x2v[v] + c2n[t] - 2.0f * acc[t][v];
        d2 = fmaxf(d2, 0.0f);
        float l = -sqrtf(d2);             // logit; l in [-~20, 0] -> exp safe
        esum[v] += __expf(l);
        if (l > bestl[v]) { bestl[v] = l; bestk[v] = k; }
      }
    }

    __syncthreads();   // all waves done reading this buffer before it is refilled
  }

  // ---- combine 16 columns per half-wave; argmax ties -> smaller k (jnp.argmax)
  #pragma unroll
  for (int v = 0; v < 8; ++v) {
    float l = bestl[v]; int k = bestk[v]; float s = esum[v];
    #pragma unroll
    for (int off = 8; off; off >>= 1) {
      float lo = __shfl_xor(l, off, 32);
      int   ko = __shfl_xor(k, off, 32);
      float so = __shfl_xor(s, off, 32);
      s += so;
      if (lo > l || (lo == l && ko < k)) { l = lo; k = ko; }
    }
    if (lq == v) {
      int rl = wave * 16 + hi * 8 + v;    // local row 0..127
      float y  = __expf(l) / s;           // y_soft at argmax
      float sc = (1.0f - y) + y;          // straight-through residual scalar
      s_label[rl] = k;
      s_scale[rl] = sc;
      labels[m0 + rl] = k;
    }
  }
  __syncthreads();

  // ---- gather: out[row] = sc * center[label]   (2 threads per row, float4 x16)
  {
    int row = tid >> 1, half = tid & 1;
    int k = s_label[row]; float sc = s_scale[row];
    const float4* srcv = reinterpret_cast<const float4*>(center + (size_t)k * D_ + half * 64);
    float4*       dstv = reinterpret_cast<float4*>(out + (size_t)(m0 + row) * D_ + half * 64);
    #pragma unroll
    for (int i = 0; i < 16; ++i) {
      float4 vv = srcv[i];
      vv.x *= sc; vv.y *= sc; vv.z *= sc; vv.w *= sc;
      dstv[i] = vv;
    }
  }
}

extern "C" void kernel_launch(void* const* d_in, const int* in_sizes, int n_in,
                              void* d_out, int out_size, void* d_ws, size_t ws_size,
                              hipStream_t stream) {
  const float* x      = (const float*)d_in[0];   // [B, D] f32
  const float* center = (const float*)d_in[1];   // [K, D] f32

  float* out         = (float*)d_out;                               // [B, D]
  float* center_copy = out + (size_t)B_ * D_;                       // [K, D]
  int*   labels      = (int*)(out + (size_t)B_ * D_ + (size_t)K_ * D_); // [B]

  float* c2 = (float*)d_ws;                                         // [K] scratch

  c2_kernel<<<K_ / 8, 256, 0, stream>>>(center, c2, center_copy);
  knn_kernel<<<B_ / 128, 256, 0, stream>>>(x, center, c2, out, labels);
}